// GraphSAGELayer_549755814532
// MI455X (gfx1250) — compile-verified
//
#include <hip/hip_runtime.h>
#include <hip/hip_bf16.h>

// GraphSAGE layer for MI455X (gfx1250, wave32).
//   neigh = segment_mean(x[edge_src] -> edge_dst)        (scatter, f32 atomics)
//   out   = [x | neigh] @ W + b                          (WMMA f32 16x16x4)
//
// Roofline: scatter moves ~1.6 GB (gather of x[src] + atomic f32 scatter) ->
// ~70-100 us at 23.3 TB/s HBM; the GEMM's 6.55 GFLOP is noise on the matrix
// pipe. So: keep fp32 exactness end-to-end, ride V_WMMA_F32_16X16X4_F32 for
// the matmul, and fuse the 1/max(deg,1) normalization into the GEMM's A-tile
// staging to avoid an extra ~100 MB pass over `neigh`.

typedef __attribute__((ext_vector_type(2))) float v2f;
typedef __attribute__((ext_vector_type(8))) float v8f;

#define IN_DIM   128
#define K_DIM    256
#define OUT_DIM  128
#define MBLK     64
#define LDS_STR  260   // 256 + 4 pad: rows land 4 banks apart -> conflict-free frag reads

// ---------------------------------------------------------------------------
// Phase 1: one wave per edge; 32 lanes x float4 = 128 features.
//   deg[dst]   += 1          (lane 0)
//   neigh[dst] += x[src]     (4 x global_atomic_add_f32 per lane)
// ---------------------------------------------------------------------------
__global__ __launch_bounds__(256) void sage_scatter(
    const float* __restrict__ x,
    const int*   __restrict__ esrc,
    const int*   __restrict__ edst,
    float*       __restrict__ neigh,
    float*       __restrict__ deg,
    int nEdges)
{
    const int wave = threadIdx.x >> 5;
    const int lane = threadIdx.x & 31;
    const int e    = blockIdx.x * 8 + wave;
    if (e >= nEdges) return;

    const int s = esrc[e];
    const int d = edst[e];
    if (lane == 0) atomicAdd(&deg[d], 1.0f);

    const float4 v = ((const float4*)(x + (size_t)s * IN_DIM))[lane];
    float* nd = neigh + (size_t)d * IN_DIM + lane * 4;
    atomicAdd(nd + 0, v.x);
    atomicAdd(nd + 1, v.y);
    atomicAdd(nd + 2, v.z);
    atomicAdd(nd + 3, v.w);
}

// ---------------------------------------------------------------------------
// Phase 2: out = [x | neigh/max(deg,1)] @ W + b
// Block = 256 threads (8 waves), M-block = 64 rows, K = 256, N = 128.
// Each wave owns a 16-wide N slice and 4 stacked 16x16 accumulators, so each
// W fragment (L2-resident) is reused 4x. A-block staged in padded LDS with the
// deg normalization applied inline.
// ---------------------------------------------------------------------------
__global__ __launch_bounds__(256) void sage_gemm(
    const float* __restrict__ x,
    const float* __restrict__ neigh,
    const float* __restrict__ deg,
    const float* __restrict__ W,      // [256][128] row-major (K-major)
    const float* __restrict__ bias,   // [128]
    float*       __restrict__ out,    // [nNodes][128]
    int nNodes)
{
    __shared__ float As[MBLK * LDS_STR];   // 64*260*4 = 66560 B

    const int mBase = blockIdx.x * MBLK;
    const int t = threadIdx.x;
    const bool fullBlock = (mBase + MBLK) <= nNodes;   // 1562 of 1563 blocks

    // ---- Stage A-block: 64 rows x 64 float4-cols = 4096 float4, 16/thread.
    // Cols 0..31 from x, cols 32..63 from neigh * (1/max(deg,1)).
    // Tail rows (>= nNodes) zero-filled so the WMMA loop runs EXEC=all-ones.
    #pragma unroll
    for (int i = 0; i < 16; ++i) {
        const int idx  = t + i * 256;     // 0..4095
        const int row  = idx >> 6;        // 0..63
        const int c4   = idx & 63;        // float4 column
        const int grow = mBase + row;
        float4 v = make_float4(0.f, 0.f, 0.f, 0.f);
        if (fullBlock || grow < nNodes) {
            if (c4 < 32) {
                v = ((const float4*)(x + (size_t)grow * IN_DIM))[c4];
            } else {
                v = ((const float4*)(neigh + (size_t)grow * IN_DIM))[c4 - 32];
                const float inv = 1.0f / fmaxf(deg[grow], 1.0f);
                v.x *= inv; v.y *= inv; v.z *= inv; v.w *= inv;
            }
        }
        float* dst = &As[row * LDS_STR + c4 * 4];
        dst[0] = v.x; dst[1] = v.y; dst[2] = v.z; dst[3] = v.w;
    }
    __syncthreads();

    const int wave  = t >> 5;        // 0..7 -> N slice
    const int lane  = t & 31;
    const int ln    = lane & 15;     // N within slice / M row within tile
    const int hi    = lane >> 4;     // K sub-pair select (A/B), M+8 select (C/D)
    const int nBase = wave * 16;

    v8f acc[4] = {};                 // 4 stacked M-tiles (64 rows total)

    for (int k0 = 0; k0 < K_DIM; k0 += 4) {
        const int kk = k0 + hi * 2;

        // B fragment: 4x16 f32 slice of W. Lanes 0-15: K={k0,k0+1}, N=ln;
        // lanes 16-31: K={k0+2,k0+3}. 16 KB/wave slice -> L0/L2 resident.
        v2f bfrag;
        bfrag.x = W[(size_t)kk       * OUT_DIM + nBase + ln];
        bfrag.y = W[(size_t)(kk + 1) * OUT_DIM + nBase + ln];

        #pragma unroll
        for (int ms = 0; ms < 4; ++ms) {
            const int row = ms * 16 + ln;
            // A fragment: lanes 0-15 row=ln K={k0,k0+1}; lanes 16-31 K+2.
            // Padded stride => 32 distinct banks across the wave.
            v2f afrag;
            afrag.x = As[row * LDS_STR + kk];
            afrag.y = As[row * LDS_STR + kk + 1];

            acc[ms] = __builtin_amdgcn_wmma_f32_16x16x4_f32(
                /*neg_a=*/false, afrag, /*neg_b=*/false, bfrag,
                /*c_mod=*/(short)0, acc[ms],
                /*reuse_a=*/false, /*reuse_b=*/false);
        }
    }

    // ---- Epilogue: D layout -> VGPR r holds (M = r + hi*8, N = ln).
    // Fast path (all but the single tail block): branch-free stores off one
    // precomputed base with compile-time immediate offsets.
    const float bv = bias[nBase + ln];
    float* obase = out + (size_t)(mBase + hi * 8) * OUT_DIM + nBase + ln;
    if (fullBlock) {
        #pragma unroll
        for (int ms = 0; ms < 4; ++ms) {
            #pragma unroll
            for (int r = 0; r < 8; ++r) {
                obase[(size_t)(ms * 16 + r) * OUT_DIM] = acc[ms][r] + bv;
            }
        }
    } else {
        #pragma unroll
        for (int ms = 0; ms < 4; ++ms) {
            #pragma unroll
            for (int r = 0; r < 8; ++r) {
                const int grow = mBase + ms * 16 + r + hi * 8;
                if (grow < nNodes)
                    out[(size_t)grow * OUT_DIM + nBase + ln] = acc[ms][r] + bv;
            }
        }
    }
}

// ---------------------------------------------------------------------------
extern "C" void kernel_launch(void* const* d_in, const int* in_sizes, int n_in,
                              void* d_out, int out_size, void* d_ws, size_t ws_size,
                              hipStream_t stream)
{
    const float* x    = (const float*)d_in[0];
    const int*   esrc = (const int*)  d_in[1];
    const int*   edst = (const int*)  d_in[2];
    const float* W    = (const float*)d_in[3];
    const float* bias = (const float*)d_in[4];
    float*       out  = (float*)d_out;

    const int nNodes = in_sizes[0] / IN_DIM;   // 100000
    const int nEdges = in_sizes[1];            // 1600000

    // Workspace: neigh accumulator [nNodes*128] then deg [nNodes].
    float* neigh = (float*)d_ws;
    float* deg   = neigh + (size_t)nNodes * IN_DIM;
    const size_t zero_bytes = ((size_t)nNodes * IN_DIM + (size_t)nNodes) * sizeof(float);
    hipMemsetAsync(d_ws, 0, zero_bytes, stream);

    const int scatterBlocks = (nEdges + 7) / 8;            // 8 edges (waves) / block
    sage_scatter<<<scatterBlocks, 256, 0, stream>>>(x, esrc, edst, neigh, deg, nEdges);

    const int gemmBlocks = (nNodes + MBLK - 1) / MBLK;     // 1563
    sage_gemm<<<gemmBlocks, 256, 0, stream>>>(x, neigh, deg, W, bias, out, nNodes);
}